// GCN_64725157151108
// MI455X (gfx1250) — compile-verified
//
#include <hip/hip_runtime.h>
#include <hip/hip_bf16.h>
#include <math.h>

typedef __attribute__((ext_vector_type(2))) float v2f;
typedef __attribute__((ext_vector_type(8))) float v8f;

#define DFEAT 64

// ---------------------------------------------------------------------------
// 1) in-degree accumulation: deg[dst[e]] += 1.0f
// ---------------------------------------------------------------------------
__global__ void gcn_deg_kernel(const int* __restrict__ dst,
                               float* __restrict__ deg, int E) {
    int e = blockIdx.x * blockDim.x + threadIdx.x;
    if (e < E) {
        __hip_atomic_fetch_add(&deg[dst[e]], 1.0f,
                               __ATOMIC_RELAXED, __HIP_MEMORY_SCOPE_AGENT);
    }
}

// ---------------------------------------------------------------------------
// 2) deg -> inv_deg in place (zero-in-degree nodes get 0, like the reference)
// ---------------------------------------------------------------------------
__global__ void gcn_invdeg_kernel(float* __restrict__ deg, int N) {
    int n = blockIdx.x * blockDim.x + threadIdx.x;
    if (n < N) {
        float d = deg[n];
        deg[n] = d > 0.0f ? 1.0f / d : 0.0f;
    }
}

// ---------------------------------------------------------------------------
// 3) edge scatter: agg[dst[e]][:] += x[src[e]][:]
//    16 threads per edge, float4 per thread, relaxed f32 atomics (resolve @L2)
// ---------------------------------------------------------------------------
__global__ void gcn_scatter_kernel(const float* __restrict__ x,
                                   const int* __restrict__ src,
                                   const int* __restrict__ dst,
                                   float* __restrict__ agg, int E) {
    int t = blockIdx.x * blockDim.x + threadIdx.x;
    int e = t >> 4;
    if (e >= E) return;
    int part = (t & 15) * 4;
    const float4 v = *(const float4*)(x + (size_t)src[e] * DFEAT + part);
    float* o = agg + (size_t)dst[e] * DFEAT + part;
    __hip_atomic_fetch_add(o + 0, v.x, __ATOMIC_RELAXED, __HIP_MEMORY_SCOPE_AGENT);
    __hip_atomic_fetch_add(o + 1, v.y, __ATOMIC_RELAXED, __HIP_MEMORY_SCOPE_AGENT);
    __hip_atomic_fetch_add(o + 2, v.z, __ATOMIC_RELAXED, __HIP_MEMORY_SCOPE_AGENT);
    __hip_atomic_fetch_add(o + 3, v.w, __ATOMIC_RELAXED, __HIP_MEMORY_SCOPE_AGENT);
}

// ---------------------------------------------------------------------------
// 4) fused (agg * inv_deg) @ W.T + b, ELU  via V_WMMA_F32_16X16X4_F32
//    Block = 128 threads = 4 waves; each wave owns one 16x16 output tile:
//    rows [blockIdx.x*16, +16), cols [wave*16, +16). D=64 -> 16 k-steps.
//
//    f32 WMMA operand layouts (ISA 7.12.2):
//      A 16x4 : lane m=lane%16 is matrix row; VGPR0={K=0|K=2}, VGPR1={K=1|K=3}
//               selected by lane half -> per lane float2 at k = kb + 2*(lane/16)
//      B 4x16 : lane n=lane%16 is matrix col; same K-pair selection.
//               B[k][n] = W[colBase+n][k]  (out = agg @ W.T)
//      C/D    : 8 VGPRs, row = r + 8*(lane/16), col = lane%16
// ---------------------------------------------------------------------------
__global__ __launch_bounds__(128)
void gcn_wmma_kernel(const float* __restrict__ agg,
                     const float* __restrict__ inv_deg,
                     const float* __restrict__ W,
                     const float* __restrict__ bias,
                     float* __restrict__ out) {
    const int lane    = threadIdx.x & 31;
    const int wave    = threadIdx.x >> 5;   // 0..3 : column tile
    const int rowBase = blockIdx.x << 4;
    const int colBase = wave << 4;
    const int m  = lane & 15;
    const int hi = lane >> 4;               // 0 or 1
    const int k2 = hi << 1;                 // K offset within each 4-step

    const float  inv  = inv_deg[rowBase + m];
    const float* arow = agg + (size_t)(rowBase + m) * DFEAT;
    const float* brow = W   + (size_t)(colBase + m) * DFEAT;

    v8f c = {};
#pragma unroll
    for (int kb = 0; kb < DFEAT; kb += 4) {
        v2f a = *(const v2f*)(arow + kb + k2);
        a.x *= inv;
        a.y *= inv;
        v2f b = *(const v2f*)(brow + kb + k2);
        // D = A x B + C   (f32 in / f32 accumulate, matches reference exactly)
        c = __builtin_amdgcn_wmma_f32_16x16x4_f32(
                /*neg_a=*/false, a, /*neg_b=*/false, b,
                /*c_mod=*/(short)0, c, /*reuse_a=*/false, /*reuse_b=*/false);
    }

    const float bcol = bias[colBase + m];
#pragma unroll
    for (int r = 0; r < 8; ++r) {
        const int row = rowBase + r + (hi << 3);
        float v = c[r] + bcol;
        v = v > 0.0f ? v : expm1f(v);       // ELU (alpha = 1)
        out[(size_t)row * DFEAT + colBase + m] = v;
    }
}

// ---------------------------------------------------------------------------
// launcher
// inputs: 0:h [N,64] f32  1:src [E] int  2:dst [E] int
//         3:W1 [64,64]    4:b1 [64]      5:W2 [64,64]   6:b2 [64]
// ws: [deg N f32][agg N*64 f32][h1 N*64 f32]  ~= 51.6 MB
// ---------------------------------------------------------------------------
extern "C" void kernel_launch(void* const* d_in, const int* in_sizes, int n_in,
                              void* d_out, int out_size, void* d_ws, size_t ws_size,
                              hipStream_t stream) {
    const float* h   = (const float*)d_in[0];
    const int*   src = (const int*)  d_in[1];
    const int*   dst = (const int*)  d_in[2];
    const float* W1  = (const float*)d_in[3];
    const float* b1  = (const float*)d_in[4];
    const float* W2  = (const float*)d_in[5];
    const float* b2  = (const float*)d_in[6];
    float*       out = (float*)d_out;

    const int N = in_sizes[0] / DFEAT;   // 100000 (divisible by 16)
    const int E = in_sizes[1];           // 1600000

    char* ws = (char*)d_ws;
    float* deg = (float*)ws;
    size_t degBytes = (((size_t)N * sizeof(float)) + 255) & ~(size_t)255;
    float* agg = (float*)(ws + degBytes);
    float* h1  = agg + (size_t)N * DFEAT;

    const int tiles    = N / 16;                 // 6250 row tiles
    const int scatterB = (E * 16 + 255) / 256;   // 16 threads / edge

    // degree + inverse
    hipMemsetAsync(deg, 0, (size_t)N * sizeof(float), stream);
    gcn_deg_kernel<<<(E + 255) / 256, 256, 0, stream>>>(dst, deg, E);
    gcn_invdeg_kernel<<<(N + 255) / 256, 256, 0, stream>>>(deg, N);

    // layer 1: scatter h -> agg, then WMMA (scale, @W1.T, +b1, ELU) -> h1
    hipMemsetAsync(agg, 0, (size_t)N * DFEAT * sizeof(float), stream);
    gcn_scatter_kernel<<<scatterB, 256, 0, stream>>>(h, src, dst, agg, E);
    gcn_wmma_kernel<<<tiles, 128, 0, stream>>>(agg, deg, W1, b1, h1);

    // layer 2: reuse agg scratch
    hipMemsetAsync(agg, 0, (size_t)N * DFEAT * sizeof(float), stream);
    gcn_scatter_kernel<<<scatterB, 256, 0, stream>>>(h1, src, dst, agg, E);
    gcn_wmma_kernel<<<tiles, 128, 0, stream>>>(agg, deg, W2, b2, out);
}